// QueryAndGroup_57827439673672
// MI455X (gfx1250) — compile-verified
//
#include <hip/hip_runtime.h>

// Problem constants (match reference setup_inputs / globals)
#define BATCH   8
#define NPTS    20000      // N
#define NPOINT  1024
#define NSAMPLE 64
#define NCH     256        // feature channels
#define CTOT    259        // 3 + NCH
#define RADIUS  0.08f
#define RAD2    (RADIUS * RADIUS)

typedef __attribute__((ext_vector_type(2))) float v2f;
typedef __attribute__((ext_vector_type(8))) float v8f;

// ---------------------------------------------------------------------------
// Kernel 1: ball query via V_WMMA_F32_16X16X4_F32, software-pipelined.
// One wave32 per block; each wave owns a tile of 16 query points and sweeps
// all NPTS points in chunks of 16.  A is the 16x4 query-coordinate matrix
// pre-scaled by -2; C is preloaded with |q|^2 + |p|^2, so the WMMA output is
// the squared distance d2 = |q|^2 + |p|^2 - 2 q.p directly.
//
// A layout (ISA 7.12.2, 32-bit A 16x4): lane L holds row M = L%16;
//   VGPR0 = K0 (lanes 0-15) / K2 (lanes 16-31); VGPR1 = K1 / K3(pad).
// B layout (4x16): lane L holds column N = L%16;
//   VGPR0 = K0 / K2; VGPR1 = K1 / K3(pad).
// D layout: VGPR k -> rows k (lanes 0-15) and k+8 (lanes 16-31), N = L%16.
//
// The next chunk's coordinates are loaded (unconditionally, clamped index)
// BEFORE the current chunk's WMMA + selection, hiding load latency behind
// ~30 VALU/WMMA instructions despite only one wave per block.
// ---------------------------------------------------------------------------
__global__ __launch_bounds__(32)
void ball_query_wmma(const float* __restrict__ xyz,
                     const float* __restrict__ new_xyz,
                     int* __restrict__ idx_out) {
  const int lane = threadIdx.x;                 // 0..31 (wave32)
  const int b    = blockIdx.x >> 6;             // 64 query-tiles per batch
  const int p0   = (blockIdx.x & 63) << 4;      // first query of this tile

  __shared__ int cnt[16];
  __shared__ int sidx[16 * NSAMPLE];

  if (lane < 16) cnt[lane] = 0;
  __syncthreads();

  // --- load this tile's query coordinates (row = lane % 16) ---
  const int q = p0 + (lane & 15);
  const float* qp = new_xyz + ((size_t)b * NPOINT + q) * 3;
  const float qx = qp[0], qy = qp[1], qz = qp[2];
  const float qn = qx * qx + qy * qy + qz * qz;

  // A matrix, pre-multiplied by -2 (loop invariant)
  v2f A;
  A.x = (lane < 16) ? (-2.0f * qx) : (-2.0f * qz);
  A.y = (lane < 16) ? (-2.0f * qy) : 0.0f;

  // Per-D-VGPR query norms: VGPR k needs qn[k] (lanes 0-15) / qn[k+8] (16-31)
  float vqn[8];
#pragma unroll
  for (int k = 0; k < 8; ++k) {
    const float lo = __shfl(qn, k, 32);
    const float hi = __shfl(qn, k + 8, 32);
    vqn[k] = (lane < 16) ? lo : hi;
  }

  const float* xb = xyz + (size_t)b * NPTS * 3;

  // --- software pipeline prologue: load chunk 0 coords ---
  const int l15 = lane & 15;
  float px = xb[l15 * 3 + 0];
  float py = xb[l15 * 3 + 1];
  float pz = xb[l15 * 3 + 2];

  for (int n0 = 0; n0 < NPTS; n0 += 16) {
    // ---- issue next chunk's loads first (clamped, unconditional) ----
    const int nn0  = (n0 + 16 < NPTS) ? (n0 + 16) : 0;
    const int nidx = nn0 + l15;
    const float nx = xb[nidx * 3 + 0];
    const float ny = xb[nidx * 3 + 1];
    const float nz = xb[nidx * 3 + 2];

    // ---- process current chunk ----
    const int pidx = n0 + l15;
    const float pn = px * px + py * py + pz * pz;

    v2f Bm;
    Bm.x = (lane < 16) ? px : pz;
    Bm.y = (lane < 16) ? py : 0.0f;

    v8f Cm;
#pragma unroll
    for (int k = 0; k < 8; ++k) Cm[k] = vqn[k] + pn;

    // D = (-2*A) x B + (|q|^2 + |p|^2)  ==  squared distance tile
    v8f D = __builtin_amdgcn_wmma_f32_16x16x4_f32(
        false, A, false, Bm, (short)0, Cm, false, false);

    // Deterministic ascending-index selection: ballot + popc prefix sum.
#pragma unroll
    for (int k = 0; k < 8; ++k) {
      const bool in = D[k] < RAD2;
      const unsigned mask = (unsigned)__ballot(in);   // uniform across wave
      if (mask != 0u) {                               // scalar skip (common)
        const unsigned half = (lane < 16) ? (mask & 0xffffu) : (mask >> 16);
        const int row  = k + ((lane >> 4) << 3);      // k or k+8
        const int base = cnt[row];                    // same addr per half
        if (in) {
          const int rank = __popc(half & ((1u << l15) - 1u));
          const int pos  = base + rank;
          if (pos < NSAMPLE) sidx[row * NSAMPLE + pos] = pidx;
        }
        if (l15 == 0) cnt[row] = base + __popc(half);
      }
    }

    // ---- rotate pipeline registers ----
    px = nx; py = ny; pz = nz;
  }
  __syncthreads();

  // Emit indices, sentinel NPTS beyond the valid count.
  int* dst = idx_out + ((size_t)b * NPOINT + p0) * NSAMPLE;
  for (int t = lane; t < 16 * NSAMPLE; t += 32) {
    const int row = t >> 6;
    const int s   = t & 63;
    dst[t] = (s < cnt[row]) ? sidx[t] : NPTS;
  }
}

// ---------------------------------------------------------------------------
// Kernel 2: gather + recenter + concat.  One block per (b, p); the sample
// dimension (64) is contiguous in the output so all stores are coalesced.
// Scattered feature reads are L2-resident (164 MB of features < 192 MB L2).
// Channel loop unrolled x8 -> 8 independent gathers in flight per thread.
// ---------------------------------------------------------------------------
__global__ __launch_bounds__(256)
void gather_group(const float* __restrict__ xyz,
                  const float* __restrict__ new_xyz,
                  const float* __restrict__ feat,
                  const int* __restrict__ idx,
                  float* __restrict__ out) {
  const int t  = threadIdx.x;
  const int bp = blockIdx.x;
  const int b  = bp / NPOINT;
  const int p  = bp - b * NPOINT;

  __shared__ int si[NSAMPLE];
  if (t < NSAMPLE) si[t] = idx[(size_t)bp * NSAMPLE + t];
  __syncthreads();

  const int s = t & 63;
  const int n = si[s];
  const bool valid = (n < NPTS);
  const int cq = t >> 6;                        // 0..3

  // xyz channels (sentinel gathers 0, then recenter/scale — matches ref)
  if (cq < 3) {
    const float cc = new_xyz[((size_t)b * NPOINT + p) * 3 + cq];
    const float v  = valid ? xyz[((size_t)b * NPTS + n) * 3 + cq] : 0.0f;
    out[(((size_t)b * CTOT + cq) * NPOINT + p) * NSAMPLE + s] =
        (v - cc) * (1.0f / RADIUS);
  }

  // feature channels, 4 per iteration (channel = c0 + cq), unrolled x8
  const float* fb = feat + ((size_t)b * NCH + cq) * NPTS;
  float* ob = out + (((size_t)b * CTOT + 3 + cq) * NPOINT + p) * NSAMPLE + s;
#pragma unroll 8
  for (int c0 = 0; c0 < NCH; c0 += 4) {
    const float f = valid ? fb[(size_t)c0 * NPTS + n] : 0.0f;
    ob[(size_t)c0 * NPOINT * NSAMPLE] = f;
  }
}

extern "C" void kernel_launch(void* const* d_in, const int* in_sizes, int n_in,
                              void* d_out, int out_size, void* d_ws, size_t ws_size,
                              hipStream_t stream) {
  const float* xyz     = (const float*)d_in[0];  // (8, 20000, 3)
  const float* new_xyz = (const float*)d_in[1];  // (8, 1024, 3)
  const float* feat    = (const float*)d_in[2];  // (8, 256, 20000)
  float* out = (float*)d_out;                    // (8, 259, 1024, 64)
  int* idx_ws = (int*)d_ws;                      // 8*1024*64 ints = 2 MB

  ball_query_wmma<<<BATCH * (NPOINT / 16), 32, 0, stream>>>(xyz, new_xyz, idx_ws);
  gather_group<<<BATCH * NPOINT, 256, 0, stream>>>(xyz, new_xyz, feat, idx_ws, out);
}